// SelfAttention_35124242546954
// MI455X (gfx1250) — compile-verified
//
#include <hip/hip_runtime.h>
#include <hip/hip_bf16.h>

typedef float v2f __attribute__((ext_vector_type(2)));
typedef float v8f __attribute__((ext_vector_type(8)));

#define BATCH 32
#define DIMSZ 4096
#define NHEAD 32
#define NKVH  8
#define HDIM  128
#define MAXSEQ 4096
#define NEG_FLT_MAX (-3.402823466e+38f)

// Async global->LDS B128: per-lane LDS byte offset (%0) and 64-bit global
// address (%1). Tracked by ASYNCcnt (CDNA5 ISA).
#define ASYNC_LOAD_B128(ldsOff, gaddr)                                  \
  asm volatile("global_load_async_to_lds_b128 %0, %1, off"              \
               :: "v"(ldsOff), "v"(gaddr) : "memory")

// ---------------------------------------------------------------------------
// GEMM: Out(32 x N) = A(32 x K, row-major) * W(K x N, row-major)
// One wave per 16x16 output tile, V_WMMA_F32_16X16X4_F32 along K.
// blockIdx.x = N tile, blockIdx.y = M tile (0 or 1), blockDim = 32 (one wave).
// ---------------------------------------------------------------------------
__global__ void gemm32_wmma_f32(const float* __restrict__ A,
                                const float* __restrict__ W,
                                float* __restrict__ Out,
                                int K, int N) {
  const int lane = threadIdx.x;       // 0..31
  const int half = lane >> 4;         // 0 | 1
  const int l16  = lane & 15;
  const int col  = blockIdx.x * 16 + l16;
  const int mrow = blockIdx.y * 16 + l16;

  // A operand: lane<16 reads K=k+0,k+1 of row mrow; lane>=16 reads K=k+2,k+3.
  const float* aptr = A + (size_t)mrow * K + 2 * half;
  // B operand: lane<16 reads rows k+0,k+1 at column col; lane>=16 rows k+2,k+3.
  const float* bptr = W + (size_t)(2 * half) * N + col;

  v8f c = {};
#pragma unroll 4
  for (int k = 0; k < K; k += 4) {
    v2f a = *(const v2f*)(aptr + k);
    v2f b;
    b.x = bptr[(size_t)k * N];
    b.y = bptr[(size_t)(k + 1) * N];
    c = __builtin_amdgcn_wmma_f32_16x16x4_f32(
        /*neg_a=*/false, a, /*neg_b=*/false, b,
        /*c_mod=*/(short)0, c, /*reuse_a=*/false, /*reuse_b=*/false);
  }

  // C layout: VGPR r holds row (mtile*16 + r + 8*half), column col.
  const int rbase = blockIdx.y * 16 + 8 * half;
#pragma unroll
  for (int r = 0; r < 8; ++r)
    Out[(size_t)(rbase + r) * N + col] = c[r];
}

// ---------------------------------------------------------------------------
// RoPE applied in-place to xq (32x4096) and xk (32x1024); S=1 so freqs row 0.
// ---------------------------------------------------------------------------
__global__ void rope_kernel(float* __restrict__ xq, float* __restrict__ xk,
                            const float* __restrict__ fcos,
                            const float* __restrict__ fsin) {
  const int QP = BATCH * NHEAD * (HDIM / 2);  // 65536 q pairs
  const int KP = BATCH * NKVH  * (HDIM / 2);  // 16384 k pairs
  int idx = blockIdx.x * blockDim.x + threadIdx.x;
  if (idx >= QP + KP) return;
  float* buf;
  int p;
  if (idx < QP) { buf = xq; p = idx; }
  else          { buf = xk; p = idx - QP; }
  const int i = p & (HDIM / 2 - 1);
  const float c = fcos[i], s = fsin[i];
  const float x0 = buf[2 * p], x1 = buf[2 * p + 1];
  buf[2 * p]     = x0 * c - x1 * s;
  buf[2 * p + 1] = x0 * s + x1 * c;
}

// ---------------------------------------------------------------------------
// Flash-decode attention. One workgroup per (b, kvh); 8 waves split the
// sequence; within a wave, lanes cover the 128-dim head (float4 per lane).
// K/V rows are double-buffered through LDS with ASYNC global->LDS loads so
// the 4KB-strided HBM stream runs ahead of compute. Online softmax per wave,
// cross-wave merge in LDS. Position start_pos reads k/v from workspace
// (the reference's cache update) — inputs are never mutated.
// ---------------------------------------------------------------------------
__global__ void attn_decode_kernel(const float* __restrict__ cache_k,
                                   const float* __restrict__ cache_v,
                                   const float* __restrict__ xq,
                                   const float* __restrict__ xk,
                                   const float* __restrict__ xv,
                                   const int* __restrict__ sp_ptr,
                                   float* __restrict__ attn_out) {
  const int b    = blockIdx.x >> 3;
  const int kvh  = blockIdx.x & 7;
  const int lane = threadIdx.x & 31;
  const int wave = threadIdx.x >> 5;  // 0..7

  const int sp = *sp_ptr;             // 4095
  const int L  = sp + 1;              // positions attended

  // Per-wave double buffers for K and V rows (async landing zone).
  __shared__ float kbuf[8][2][HDIM];  // 8 KB
  __shared__ float vbuf[8][2][HDIM];  // 8 KB

  // Query registers: 4 grouped heads, 4 dims per lane, pre-scaled by 1/sqrt(HD).
  const float qscale = 0.088388347648318447f;  // 1/sqrt(128)
  float qr[4][4];
#pragma unroll
  for (int h = 0; h < 4; ++h) {
    const float* qp = xq + (size_t)b * DIMSZ + (kvh * 4 + h) * HDIM + lane * 4;
    const float4 q4 = *(const float4*)qp;
    qr[h][0] = q4.x * qscale; qr[h][1] = q4.y * qscale;
    qr[h][2] = q4.z * qscale; qr[h][3] = q4.w * qscale;
  }

  float m_i[4], l_i[4], acc[4][4];
#pragma unroll
  for (int h = 0; h < 4; ++h) {
    m_i[h] = NEG_FLT_MAX; l_i[h] = 0.f;
    acc[h][0] = acc[h][1] = acc[h][2] = acc[h][3] = 0.f;
  }

  const int chunk = (L + 7) >> 3;
  const int t0 = wave * chunk;
  const int t1 = (t0 + chunk < L) ? (t0 + chunk) : L;

  // Issue async loads of K/V row t into LDS buffer `buf` (16B per lane).
  auto issue_kv = [&](int t, int buf) {
    const float* kp;
    const float* vp;
    if (t == sp) {  // new token: from workspace (cache not mutated)
      const size_t off = ((size_t)b * NKVH + kvh) * HDIM;
      kp = xk + off; vp = xv + off;
    } else {
      const size_t off = (((size_t)b * MAXSEQ + t) * NKVH + kvh) * HDIM;
      kp = cache_k + off; vp = cache_v + off;
    }
    // Low 32 bits of a generic pointer to __shared__ == LDS byte offset.
    const unsigned kOff =
        (unsigned)(unsigned long long)(uintptr_t)&kbuf[wave][buf][lane * 4];
    const unsigned vOff =
        (unsigned)(unsigned long long)(uintptr_t)&vbuf[wave][buf][lane * 4];
    const unsigned long long kG = (unsigned long long)(uintptr_t)(kp + lane * 4);
    const unsigned long long vG = (unsigned long long)(uintptr_t)(vp + lane * 4);
    ASYNC_LOAD_B128(kOff, kG);
    ASYNC_LOAD_B128(vOff, vG);
  };

  int buf = 0;
  if (t0 < t1) issue_kv(t0, buf);

  for (int t = t0; t < t1; ++t) {
    const bool has_next = (t + 1 < t1);
    if (has_next) {
      issue_kv(t + 1, buf ^ 1);
      // 2 ops for t+1 outstanding; loads complete in order, so <=2 means
      // buffer `buf` (row t) is fully resident in LDS.
      asm volatile("s_wait_asynccnt 0x2" ::: "memory");
    } else {
      asm volatile("s_wait_asynccnt 0x0" ::: "memory");
    }

    const float4 kv = *(const float4*)&kbuf[wave][buf][lane * 4];
    const float4 vv = *(const float4*)&vbuf[wave][buf][lane * 4];

    float s[4];
#pragma unroll
    for (int h = 0; h < 4; ++h)
      s[h] = qr[h][0] * kv.x + qr[h][1] * kv.y + qr[h][2] * kv.z + qr[h][3] * kv.w;

    // Butterfly reduce across the wave: every lane ends with the full dot.
#pragma unroll
    for (int off = 16; off >= 1; off >>= 1) {
#pragma unroll
      for (int h = 0; h < 4; ++h) s[h] += __shfl_xor(s[h], off, 32);
    }

#pragma unroll
    for (int h = 0; h < 4; ++h) {
      const float mn    = fmaxf(m_i[h], s[h]);
      const float scale = __expf(m_i[h] - mn);
      const float e     = __expf(s[h] - mn);
      l_i[h] = l_i[h] * scale + e;
      acc[h][0] = acc[h][0] * scale + e * vv.x;
      acc[h][1] = acc[h][1] * scale + e * vv.y;
      acc[h][2] = acc[h][2] * scale + e * vv.z;
      acc[h][3] = acc[h][3] * scale + e * vv.w;
      m_i[h] = mn;
    }
    buf ^= 1;
  }

  // Cross-wave merge in LDS.
  __shared__ float red_m[8][4];
  __shared__ float red_l[8][4];
  __shared__ float red_acc[8][4][HDIM];  // 16 KB
  if (lane == 0) {
#pragma unroll
    for (int h = 0; h < 4; ++h) { red_m[wave][h] = m_i[h]; red_l[wave][h] = l_i[h]; }
  }
#pragma unroll
  for (int h = 0; h < 4; ++h) {
#pragma unroll
    for (int j = 0; j < 4; ++j) red_acc[wave][h][lane * 4 + j] = acc[h][j];
  }
  __syncthreads();

  for (int o = threadIdx.x; o < 4 * HDIM; o += blockDim.x) {
    const int h = o >> 7;
    const int d = o & (HDIM - 1);
    float M = NEG_FLT_MAX;
#pragma unroll
    for (int w = 0; w < 8; ++w) M = fmaxf(M, red_m[w][h]);
    float lsum = 0.f, val = 0.f;
#pragma unroll
    for (int w = 0; w < 8; ++w) {
      const float sc = __expf(red_m[w][h] - M);
      lsum += red_l[w][h] * sc;
      val  += red_acc[w][h][d] * sc;
    }
    attn_out[(size_t)b * DIMSZ + (kvh * 4 + h) * HDIM + d] = val / lsum;
  }
}

// ---------------------------------------------------------------------------
extern "C" void kernel_launch(void* const* d_in, const int* in_sizes, int n_in,
                              void* d_out, int out_size, void* d_ws, size_t ws_size,
                              hipStream_t stream) {
  const float* x   = (const float*)d_in[0];   // (32,1,4096)
  const float* wq  = (const float*)d_in[1];   // (4096,4096)
  const float* wk  = (const float*)d_in[2];   // (4096,1024)
  const float* wv  = (const float*)d_in[3];   // (4096,1024)
  const float* wo  = (const float*)d_in[4];   // (4096,4096)
  const float* ck  = (const float*)d_in[5];   // (32,4096,8,128)
  const float* cv  = (const float*)d_in[6];   // (32,4096,8,128)
  const float* fco = (const float*)d_in[7];   // (1,64)
  const float* fsi = (const float*)d_in[8];   // (1,64)
  const int*   sp  = (const int*)d_in[9];     // scalar

  float* ws   = (float*)d_ws;
  float* xq   = ws;                                   // 32*4096
  float* xk   = xq + BATCH * DIMSZ;                   // 32*1024
  float* xv   = xk + BATCH * NKVH * HDIM;             // 32*1024
  float* attn = xv + BATCH * NKVH * HDIM;             // 32*4096

  // QKV projections (WMMA f32): grid = (N/16, 2 M-tiles), one wave per tile.
  gemm32_wmma_f32<<<dim3(DIMSZ / 16, 2), 32, 0, stream>>>(x, wq, xq, DIMSZ, DIMSZ);
  gemm32_wmma_f32<<<dim3(NKVH * HDIM / 16, 2), 32, 0, stream>>>(x, wk, xk, DIMSZ, NKVH * HDIM);
  gemm32_wmma_f32<<<dim3(NKVH * HDIM / 16, 2), 32, 0, stream>>>(x, wv, xv, DIMSZ, NKVH * HDIM);

  // RoPE on xq and xk (in workspace).
  const int pairs = BATCH * (NHEAD + NKVH) * (HDIM / 2);  // 81920
  rope_kernel<<<(pairs + 255) / 256, 256, 0, stream>>>(xq, xk, fco, fsi);

  // Flash-decode attention: one WG per (b, kvh), async KV staging.
  attn_decode_kernel<<<BATCH * NKVH, 256, 0, stream>>>(ck, cv, xq, xk, xv, sp, attn);

  // Output projection into d_out.
  gemm32_wmma_f32<<<dim3(DIMSZ / 16, 2), 32, 0, stream>>>(attn, wo, (float*)d_out, DIMSZ, DIMSZ);
}